// GCNModel_44710609551614
// MI455X (gfx1250) — compile-verified
//
#include <hip/hip_runtime.h>
#include <hip/hip_bf16.h>

typedef __attribute__((ext_vector_type(2))) float v2f;
typedef __attribute__((ext_vector_type(8))) float v8f;

// ---------------- utility kernels ----------------

__global__ void fill_f32(float* __restrict__ p, float v, int n) {
    int i = blockIdx.x * blockDim.x + threadIdx.x;
    if (i < n) p[i] = v;
}

// deg[dst[e]] += w[e]   (deg pre-initialized to 1.0 for the self-loop weight)
__global__ void deg_accum(const long long* __restrict__ dst,
                          const float* __restrict__ w,
                          float* __restrict__ deg, int E) {
    int e = blockIdx.x * blockDim.x + threadIdx.x;
    if (e < E) atomicAdd(&deg[(int)dst[e]], w[e]);
}

// in-place: deg -> deg^-1/2 (0 if deg<=0)
__global__ void dinv_kernel(float* __restrict__ deg, int n) {
    int i = blockIdx.x * blockDim.x + threadIdx.x;
    if (i < n) {
        float d = deg[i];
        deg[i] = d > 0.f ? rsqrtf(d) : 0.f;
    }
}

// norm[e] = dinv[src]*w*dinv[dst] for real edges; dinv[i]^2 for self-loops
__global__ void norm_kernel(const long long* __restrict__ src,
                            const long long* __restrict__ dst,
                            const float* __restrict__ w,
                            const float* __restrict__ dinv,
                            float* __restrict__ nrm, int E, int n) {
    int e = blockIdx.x * blockDim.x + threadIdx.x;
    if (e < E) {
        nrm[e] = dinv[(int)src[e]] * w[e] * dinv[(int)dst[e]];
    } else if (e < E + n) {
        float d = dinv[e - E];
        nrm[e] = d * d;
    }
}

// ---------------- WMMA fp32 GEMM: C[M x NDIM] = A[M x KDIM] * B[KDIM x NDIM] ----------------
// 256 threads = 8 waves per block; each wave owns a 16-row stripe and all NDIM/16
// column tiles, accumulating with V_WMMA_F32_16X16X4_F32 over K in steps of 4.
// M is a multiple of 16, so tile exit is wave-uniform and EXEC stays all-1s.

template <int KDIM, int NDIM>
__global__ __launch_bounds__(256) void gemm_wmma_f32(const float* __restrict__ A,
                                                     const float* __restrict__ B,
                                                     float* __restrict__ C, int M) {
    constexpr int NT = NDIM / 16;
    const int wave = threadIdx.x >> 5;
    const int lane = threadIdx.x & 31;
    const int half = lane >> 4;   // 0: lanes 0-15, 1: lanes 16-31
    const int l16  = lane & 15;
    const int r = (blockIdx.x * 8 + wave) * 16;   // first output row of this wave
    if (r >= M) return;                            // wave-uniform exit

    v8f acc[NT];
#pragma unroll
    for (int n = 0; n < NT; ++n) acc[n] = (v8f){0.f,0.f,0.f,0.f,0.f,0.f,0.f,0.f};

    // A-fragment base: lane holds row (r + l16), K offset 2*half (ISA 16x4 f32 layout)
    const float* arow = A + (size_t)(r + l16) * KDIM + 2 * half;

    for (int k = 0; k < KDIM; k += 4) {
        v2f a = *(const v2f*)(arow + k);   // {A[row][k+2h], A[row][k+2h+1]} (8B aligned)
#pragma unroll
        for (int n = 0; n < NT; ++n) {
            v2f b;
            b.x = B[(size_t)(k + 2 * half)     * NDIM + n * 16 + l16];
            b.y = B[(size_t)(k + 2 * half + 1) * NDIM + n * 16 + l16];
            acc[n] = __builtin_amdgcn_wmma_f32_16x16x4_f32(
                /*neg_a=*/false, a, /*neg_b=*/false, b,
                /*c_mod=*/(short)0, acc[n], /*reuse_a=*/false, /*reuse_b=*/false);
        }
    }

    // C/D layout: VGPR j -> row j + 8*half, col l16
#pragma unroll
    for (int n = 0; n < NT; ++n) {
#pragma unroll
        for (int j = 0; j < 8; ++j) {
            C[(size_t)(r + j + 8 * half) * NDIM + n * 16 + l16] = acc[n][j];
        }
    }
}

// ---------------- edge-parallel scatter-add: out[dst] += P[src] * norm ----------------
// One wave32 per edge; lanes stride the feature dimension (coalesced f32 atomics).

template <int F>
__global__ __launch_bounds__(256) void scatter_agg(const long long* __restrict__ srcI,
                                                   const long long* __restrict__ dstI,
                                                   const float* __restrict__ nrm,
                                                   const float* __restrict__ P,
                                                   float* __restrict__ out, int E, int n) {
    const int e = blockIdx.x * (blockDim.x >> 5) + (threadIdx.x >> 5);
    const int lane = threadIdx.x & 31;
    if (e >= E + n) return;
    int s, d;
    if (e < E) { s = (int)srcI[e]; d = (int)dstI[e]; }
    else       { s = e - E;        d = s;            }
    const float w = nrm[e];
    const float* prow = P + (size_t)s * F;
    float* orow = out + (size_t)d * F;
    __builtin_prefetch(prow, 0, 1);   // global_prefetch on the gathered row
#pragma unroll
    for (int f = lane; f < F; f += 32)
        atomicAdd(&orow[f], prow[f] * w);
}

// ---------------- bias + relu (in place) ----------------

template <int F>
__global__ void bias_relu(float* __restrict__ x, const float* __restrict__ b, int n) {
    int i = blockIdx.x * blockDim.x + threadIdx.x;
    if (i < n * F) {
        float v = x[i] + b[i % F];
        x[i] = v > 0.f ? v : 0.f;
    }
}

// ---------------- launch ----------------

extern "C" void kernel_launch(void* const* d_in, const int* in_sizes, int n_in,
                              void* d_out, int out_size, void* d_ws, size_t ws_size,
                              hipStream_t stream) {
    const float*     x  = (const float*)d_in[0];
    const long long* ei = (const long long*)d_in[1];   // int64 [2, E]
    const float*     ew = (const float*)d_in[2];
    const float*     W1 = (const float*)d_in[3];
    const float*     b1 = (const float*)d_in[4];
    const float*     W2 = (const float*)d_in[5];
    const float*     b2 = (const float*)d_in[6];
    float*           out = (float*)d_out;

    constexpr int IN = 256, H = 96, OUT = 64;
    const int N = in_sizes[0] / IN;
    const int E = in_sizes[2];
    const long long* src = ei;
    const long long* dst = ei + E;

    float* ws   = (float*)d_ws;
    float* deg  = ws;                      // N floats (becomes dinv)
    float* nrm  = deg + N;                 // E+N floats
    float* bufA = nrm + (E + N);           // N*H floats (P1, then reused for P2)
    float* bufB = bufA + (size_t)N * H;    // N*H floats (agg1 -> h)

    const int T = 256;

    // re-initialize accumulators every call (harness does not re-zero between replays)
    fill_f32<<<(N + T - 1) / T, T, 0, stream>>>(deg, 1.0f, N);           // self-loop weight
    fill_f32<<<(N * H + T - 1) / T, T, 0, stream>>>(bufB, 0.0f, N * H);
    fill_f32<<<(N * OUT + T - 1) / T, T, 0, stream>>>(out, 0.0f, N * OUT);

    // symmetric normalization
    deg_accum <<<(E + T - 1) / T, T, 0, stream>>>(dst, ew, deg, E);
    dinv_kernel<<<(N + T - 1) / T, T, 0, stream>>>(deg, N);
    norm_kernel<<<(E + N + T - 1) / T, T, 0, stream>>>(src, dst, ew, deg, nrm, E, N);

    // layer 1: P1 = x @ W1 ; agg ; h = relu(agg + b1)
    gemm_wmma_f32<IN, H><<<(N + 127) / 128, 256, 0, stream>>>(x, W1, bufA, N);
    scatter_agg<H><<<(E + N + 7) / 8, 256, 0, stream>>>(src, dst, nrm, bufA, bufB, E, N);
    bias_relu<H><<<(N * H + T - 1) / T, T, 0, stream>>>(bufB, b1, N);

    // layer 2: P2 = h @ W2 ; agg into d_out ; out = relu(agg + b2)
    gemm_wmma_f32<H, OUT><<<(N + 127) / 128, 256, 0, stream>>>(bufB, W2, bufA, N);
    scatter_agg<OUT><<<(E + N + 7) / 8, 256, 0, stream>>>(src, dst, nrm, bufA, out, E, N);
    bias_relu<OUT><<<(N * OUT + T - 1) / T, T, 0, stream>>>(out, b2, N);
}